// Spring_7713761264159
// MI455X (gfx1250) — compile-verified
//
#include <hip/hip_runtime.h>

typedef __attribute__((ext_vector_type(2))) float v2f;
typedef __attribute__((ext_vector_type(8))) float v8f;

#define EPSILON 1e-5f
#define NPTS 2048
#define NTILES (NPTS / 16)   // 128 j-tiles
#define WAVES 8
#define TPB 256

__global__ __launch_bounds__(TPB)
void spring_wmma_kernel(const float* __restrict__ X,
                        const float* __restrict__ Kp,
                        const float* __restrict__ Bp,
                        float* __restrict__ Out) {
    // Whole batch staged in LDS as (x, y, z, |x|^2): 32 KB of the 320 KB WGP LDS.
    __shared__ float4 pts[NPTS];
    __shared__ float  part[WAVES][16][4];   // per-wave partial (S, Sx, Sy, Sz) per i-row

    const int tid  = threadIdx.x;
    const int lane = tid & 31;
    const int wv   = tid >> 5;
    const int half = lane >> 4;     // 0: lanes 0-15, 1: lanes 16-31
    const int l15  = lane & 15;

    const int batch = blockIdx.x >> 7;       // 128 i-tiles per batch
    const int iTile = blockIdx.x & 127;
    const int iBase = iTile * 16;

    const float Kv = Kp[0];
    const float Bv = Bp[0];

    const float* Xb = X + batch * (NPTS * 3);

    // Stage points + squared norms into LDS.
    for (int n = tid; n < NPTS; n += TPB) {
        float x = Xb[3 * n + 0];
        float y = Xb[3 * n + 1];
        float z = Xb[3 * n + 2];
        pts[n] = make_float4(x, y, z, x * x + y * y + z * z);
    }
    __syncthreads();

    // A-matrix fragment (16x4 f32): lane<16 -> (K0,K1)=(x_i,y_i); lane>=16 -> (K2,K3)=(z_i,0)
    float4 qa = pts[iBase + l15];
    v2f Afrag;
    Afrag[0] = half ? qa.z : qa.x;
    Afrag[1] = half ? 0.0f : qa.y;

    // |x_i|^2 for the 8 C/D rows this lane owns: row M = v + 8*half
    float ni[8];
#pragma unroll
    for (int v = 0; v < 8; ++v) ni[v] = pts[iBase + 8 * half + v].w;

    float accS[8], accX[8], accY[8], accZ[8];
#pragma unroll
    for (int v = 0; v < 8; ++v) { accS[v] = 0.f; accX[v] = 0.f; accY[v] = 0.f; accZ[v] = 0.f; }

    // 8 waves of this block split the 128 j-tiles.
    for (int jt = wv; jt < NTILES; jt += WAVES) {
        const int jBase = jt * 16;
        float4 p = pts[jBase + l15];          // point j = jBase + (lane&15): ds_load_b128

        // B-matrix fragment (4x16 f32): lane<16 -> (K0,K1)=(x_j,y_j); lane>=16 -> (K2,K3)=(z_j,0)
        v2f Bfrag;
        Bfrag[0] = half ? p.z : p.x;
        Bfrag[1] = half ? 0.0f : p.y;

        v8f c = {};
        // Gram tile: g[v] = <x_i, x_j> for row M = v + 8*half, col N = lane&15
        v8f g = __builtin_amdgcn_wmma_f32_16x16x4_f32(false, Afrag, false, Bfrag,
                                                      (short)0, c, false, false);
#pragma unroll
        for (int v = 0; v < 8; ++v) {
            float t = ni[v] + p.w - 2.0f * g[v];      // d^2 = |xi|^2 + |xj|^2 - 2<xi,xj>
            t = fmaxf(t, 0.0f);
            float d = __builtin_amdgcn_sqrtf(t);      // sqrt(0)=0: matches safe-sqrt diagonal
            float w = __builtin_amdgcn_rcpf(d + EPSILON);
            float s = (d - Bv) * w;                   // K folded in at the end
            accS[v] += s;
            accX[v] = fmaf(s, p.x, accX[v]);
            accY[v] = fmaf(s, p.y, accY[v]);
            accZ[v] = fmaf(s, p.z, accZ[v]);
        }
    }

    // Reduce over the 16 columns (lanes within each half); xor masks 1,2,4,8 stay inside a half.
#pragma unroll
    for (int m = 1; m < 16; m <<= 1) {
#pragma unroll
        for (int v = 0; v < 8; ++v) {
            accS[v] += __shfl_xor(accS[v], m, 32);
            accX[v] += __shfl_xor(accX[v], m, 32);
            accY[v] += __shfl_xor(accY[v], m, 32);
            accZ[v] += __shfl_xor(accZ[v], m, 32);
        }
    }

    if (l15 == 0) {
#pragma unroll
        for (int v = 0; v < 8; ++v) {
            int row = v + 8 * half;
            part[wv][row][0] = accS[v];
            part[wv][row][1] = accX[v];
            part[wv][row][2] = accY[v];
            part[wv][row][3] = accZ[v];
        }
    }
    __syncthreads();

    // Combine the 8 wave partials and write the 16 output points of this i-tile.
    if (tid < 16) {
        float S = 0.f, Sx = 0.f, Sy = 0.f, Sz = 0.f;
#pragma unroll
        for (int w2 = 0; w2 < WAVES; ++w2) {
            S  += part[w2][tid][0];
            Sx += part[w2][tid][1];
            Sy += part[w2][tid][2];
            Sz += part[w2][tid][3];
        }
        const int i = iBase + tid;
        float4 q = pts[i];
        // sum_j K*(d-B)*(x_i - x_j)/(d+eps) = K*(x_i*S - Sx)
        float ox = q.x + Kv * (q.x * S - Sx);
        float oy = q.y + Kv * (q.y * S - Sy);
        float oz = q.z + Kv * (q.z * S - Sz);
        float* o = Out + batch * (NPTS * 3) + i * 3;
        o[0] = ox;
        o[1] = oy;
        o[2] = oz;
    }
}

extern "C" void kernel_launch(void* const* d_in, const int* in_sizes, int n_in,
                              void* d_out, int out_size, void* d_ws, size_t ws_size,
                              hipStream_t stream) {
    const float* X = (const float*)d_in[0];
    const float* K = (const float*)d_in[1];
    const float* B = (const float*)d_in[2];
    float* out = (float*)d_out;

    const int batches = in_sizes[0] / (NPTS * 3);   // 4 for the reference setup
    dim3 grid(batches * NTILES);                    // one block per (batch, 16-row i-tile)
    dim3 block(TPB);
    hipLaunchKernelGGL(spring_wmma_kernel, grid, block, 0, stream, X, K, B, out);
}